// DeepSeekMoe_63625645523144
// MI455X (gfx1250) — compile-verified
//
#include <hip/hip_runtime.h>
#include <hip/hip_bf16.h>

// MoE SwiGLU, fp32, using native CDNA5 V_WMMA_F32_16X16X4_F32.
// T=4096, D=1024, DFF=1408, E=8, K=2.

#define T_TOK 4096
#define DIM   1024
#define DFF   1408
#define NEXP  8
#define TOPK  2
#define CAP   (T_TOK * TOPK)      // worst-case tokens per expert
#define MTILE 16
#define MAX_TILES (CAP / MTILE)   // 512
// padded LDS strides (floats): +4 so row-to-row delta % 256B == 16B (no bank conflicts)
#define XS_STRIDE (DIM + 4)       // 1028
#define HS_STRIDE (DFF + 4)       // 1412

typedef __attribute__((ext_vector_type(2))) float v2f;
typedef __attribute__((ext_vector_type(8))) float v8f;

__global__ void moe_zero_kernel(float* __restrict__ out, int n, int* __restrict__ cnt) {
    int i = blockIdx.x * blockDim.x + threadIdx.x;
    int stride = gridDim.x * blockDim.x;
    for (; i < n; i += stride) out[i] = 0.0f;
    if (blockIdx.x == 0 && threadIdx.x < NEXP) cnt[threadIdx.x] = 0;
}

__global__ void moe_route_kernel(const int* __restrict__ idx,
                                 const float* __restrict__ w,
                                 int* __restrict__ cnt,
                                 int* __restrict__ tok,
                                 float* __restrict__ wgt) {
    int i = blockIdx.x * blockDim.x + threadIdx.x;   // pair index over [T, K]
    if (i < T_TOK * TOPK) {
        int e = idx[i];
        int pos = atomicAdd(&cnt[e], 1);
        tok[e * CAP + pos] = i / TOPK;
        wgt[e * CAP + pos] = w[i];
    }
}

__global__ __launch_bounds__(256) void moe_expert_kernel(
    const float* __restrict__ x,
    const float* __restrict__ Wg,
    const float* __restrict__ Wu,
    const float* __restrict__ Wd,
    const int*  __restrict__ cnt,
    const int*  __restrict__ tok,
    const float* __restrict__ wgt,
    float* __restrict__ out)
{
    __shared__ float x_s[MTILE * XS_STRIDE];   // ~64.3 KB  gathered x rows
    __shared__ float h_s[MTILE * HS_STRIDE];   // ~88.3 KB  SwiGLU hidden
    __shared__ int   tok_s[MTILE];
    __shared__ float w_s[MTILE];

    const int e    = blockIdx.x / MAX_TILES;
    const int tile = blockIdx.x % MAX_TILES;
    const int n_e  = cnt[e];
    const int row0 = tile * MTILE;
    if (row0 >= n_e) return;                    // block-uniform exit

    const int tid = threadIdx.x;

    if (tid < MTILE) {
        int r = row0 + tid;
        if (r < n_e) { tok_s[tid] = tok[e * CAP + r]; w_s[tid] = wgt[e * CAP + r]; }
        else         { tok_s[tid] = -1;               w_s[tid] = 0.0f; }
    }
    __syncthreads();

    // Stage gathered x rows into LDS (float4), zeros for padding rows.
    for (int i = tid; i < MTILE * (DIM / 4); i += 256) {
        int m = i / (DIM / 4);
        int c = i % (DIM / 4);
        float4 v = make_float4(0.0f, 0.0f, 0.0f, 0.0f);
        int t = tok_s[m];
        if (t >= 0) v = ((const float4*)(x + (size_t)t * DIM))[c];
        ((float4*)(x_s + m * XS_STRIDE))[c] = v;
    }
    __syncthreads();

    const int lane  = tid & 31;
    const int wave  = tid >> 5;        // 0..7
    const int lhalf = lane >> 4;       // 0/1: K-half selector in A/B fragments
    const int lmod  = lane & 15;       // M (A) or N (B/C/D) index

    const float* Wg_e = Wg + (size_t)e * DIM * DFF;
    const float* Wu_e = Wu + (size_t)e * DIM * DFF;
    const float* Wd_e = Wd + (size_t)e * DFF * DIM;

    // ---------------- Phase 1: h = silu(x@Wg) * (x@Wu) ----------------
    for (int nt = wave; nt < DFF / 16; nt += 8) {      // 88 tiles, 11/wave: uniform
        const int n0 = nt * 16;
        v8f cg = {};
        v8f cu = {};
        const float* colg = Wg_e + n0 + lmod;
        const float* colu = Wu_e + n0 + lmod;
        const float* arow = x_s + lmod * XS_STRIDE;
        for (int k0 = 0; k0 < DIM; k0 += 4) {
            const int kb = k0 + 2 * lhalf;
            v2f a = *(const v2f*)(arow + kb);           // A[m= lmod][k=kb,kb+1]
            v2f bg, bu;
            bg[0] = colg[(size_t)kb * DFF];
            bg[1] = colg[(size_t)(kb + 1) * DFF];
            bu[0] = colu[(size_t)kb * DFF];
            bu[1] = colu[(size_t)(kb + 1) * DFF];
            cg = __builtin_amdgcn_wmma_f32_16x16x4_f32(false, a, false, bg, (short)0, cg, false, false);
            cu = __builtin_amdgcn_wmma_f32_16x16x4_f32(false, a, false, bu, (short)0, cu, false, false);
        }
#pragma unroll
        for (int r = 0; r < 8; ++r) {
            float g = cg[r];
            float s = g / (1.0f + __expf(-g));          // silu
            int m = r + 8 * lhalf;                      // C/D layout row
            h_s[m * HS_STRIDE + n0 + lmod] = s * cu[r];
        }
    }
    __syncthreads();

    // ---------------- Phase 2: y = h @ Wd, scaled accumulate ----------------
    for (int nt = wave; nt < DIM / 16; nt += 8) {      // 64 tiles, 8/wave: uniform
        const int n0 = nt * 16;
        v8f cy = {};
        const float* cold = Wd_e + n0 + lmod;
        const float* arow = h_s + lmod * HS_STRIDE;
        for (int k0 = 0; k0 < DFF; k0 += 4) {
            const int kb = k0 + 2 * lhalf;
            v2f a = *(const v2f*)(arow + kb);
            v2f b;
            b[0] = cold[(size_t)kb * DIM];
            b[1] = cold[(size_t)(kb + 1) * DIM];
            cy = __builtin_amdgcn_wmma_f32_16x16x4_f32(false, a, false, b, (short)0, cy, false, false);
        }
#pragma unroll
        for (int r = 0; r < 8; ++r) {
            int m = r + 8 * lhalf;
            int t = tok_s[m];
            if (t >= 0) {
                atomicAdd(out + (size_t)t * DIM + n0 + lmod, cy[r] * w_s[m]);
            }
        }
    }
}

extern "C" void kernel_launch(void* const* d_in, const int* in_sizes, int n_in,
                              void* d_out, int out_size, void* d_ws, size_t ws_size,
                              hipStream_t stream) {
    const float* x  = (const float*)d_in[0];
    const float* Wg = (const float*)d_in[1];
    const float* Wu = (const float*)d_in[2];
    const float* Wd = (const float*)d_in[3];
    const float* tw = (const float*)d_in[4];
    const int*   ti = (const int*)d_in[5];
    float* out = (float*)d_out;

    // Workspace layout: [cnt: 8 ints][pad to 512B][tok: 8*8192 ints][wgt: 8*8192 floats]
    int*   cnt = (int*)d_ws;
    int*   tok = (int*)((char*)d_ws + 512);
    float* wgt = (float*)((char*)d_ws + 512 + (size_t)NEXP * CAP * sizeof(int));

    moe_zero_kernel<<<1024, 256, 0, stream>>>(out, T_TOK * DIM, cnt);
    moe_route_kernel<<<(T_TOK * TOPK + 255) / 256, 256, 0, stream>>>(ti, tw, cnt, tok, wgt);
    moe_expert_kernel<<<NEXP * MAX_TILES, 256, 0, stream>>>(x, Wg, Wu, Wd, cnt, tok, wgt, out);
}